// NoRelativePositionFeatures_16587163697707
// MI455X (gfx1250) — compile-verified
//
#include <hip/hip_runtime.h>

typedef float v2f __attribute__((ext_vector_type(2)));
typedef float v4f __attribute__((ext_vector_type(4)));
typedef float v8f __attribute__((ext_vector_type(8)));

#if defined(__HIP_DEVICE_COMPILE__)
#  if __has_builtin(__builtin_amdgcn_wmma_f32_16x16x4_f32)
#    define USE_WMMA 1
#  else
#    define USE_WMMA 0
#  endif
#else
#  define USE_WMMA 0
#endif

#define NPTS 49
#define D3   42
#define EOUT 128
#define GS   68          // Gram row stride in floats (16B aligned, conflict-friendly)
#define POIS 1.0e30f     // poison for diagonal / invalid columns (d2 = SQ[i]+POIS, huge)
#define BIGF 1.0e30f

// ---------------------------------------------------------------------------
// Kernel 1: collapse the rank-2 structure.  u[e], v[e], c[e]  (3*128 floats)
// ---------------------------------------------------------------------------
__global__ __launch_bounds__(EOUT) void uvc_kernel(
    const float* __restrict__ W_dist, const float* __restrict__ b_dist,
    const float* __restrict__ emb,    const float* __restrict__ W_den,
    const float* __restrict__ b_den,  const float* __restrict__ W_out,
    const float* __restrict__ b_out,  float* __restrict__ uvc)
{
    const int e = threadIdx.x;                 // 0..127
    const float* embN = emb + NPTS * D3;       // emb[49,:]
    float u = 0.0f, v = 0.0f, c = b_out[e];
    for (int j = 0; j < D3; ++j) {
        float w0 = W_out[(j        ) * EOUT + e];
        float w1 = W_out[(D3   + j ) * EOUT + e];
        float w2 = W_out[(2*D3 + j ) * EOUT + e];
        u = fmaf(W_dist[j], w0, u);
        v = fmaf(W_den[j],  w2, v);
        c = fmaf(b_dist[j], w0, c);
        c = fmaf(embN[j],   w1, c);
        c = fmaf(b_den[j],  w2, c);
    }
    uvc[e]          = u;
    uvc[EOUT + e]   = v;
    uvc[2*EOUT + e] = c;
}

// ---------------------------------------------------------------------------
// Kernel 2: one wave per batch item.
//   - 16x V_WMMA_F32_16X16X4_F32 -> padded 64x64 Gram matrix P*P^T
//   - fused store  G'[i][j] = SQ[j] - 2*ip  to LDS (diag/invalid poisoned)
//   - each lane serially scans its own row(s): 5-op sorted-triple insert
//   - streamed rank-2 output:  out = cd*u + den*v + c  (nontemporal b128)
// ---------------------------------------------------------------------------
__global__ __launch_bounds__(32) void fused_kernel(
    const float* __restrict__ points,
    const float* __restrict__ uvc,
    float* __restrict__ out)
{
    __shared__ __align__(16) float P[64 * 4];     // padded points (x,y,z,0)
    __shared__ __align__(16) float G[64 * GS];    // fused Gram rows
    __shared__ float SQ[64];
    __shared__ __align__(8) float FD[64 * 2];     // (cd, den) pairs

    const int lane = threadIdx.x;
    const int b    = blockIdx.x;
    const float* pb = points + (size_t)b * (NPTS * 3);

    // zero padded point storage, scatter-load 49*3 floats
    ((v4f*)P)[lane]      = (v4f)0.0f;
    ((v4f*)P)[lane + 32] = (v4f)0.0f;
    __syncthreads();
    for (int idx = lane; idx < NPTS * 3; idx += 32) {
        int r = idx / 3;
        int c = idx - r * 3;
        P[r * 4 + c] = pb[idx];
    }
    __syncthreads();

    // centroid via full-wave reduction
    float sx = 0.0f, sy = 0.0f, sz = 0.0f;
    for (int r = lane; r < NPTS; r += 32) {
        sx += P[r * 4 + 0]; sy += P[r * 4 + 1]; sz += P[r * 4 + 2];
    }
    #pragma unroll
    for (int m = 16; m >= 1; m >>= 1) {
        sx += __shfl_xor(sx, m, 32);
        sy += __shfl_xor(sy, m, 32);
        sz += __shfl_xor(sz, m, 32);
    }
    const float inv_n = 1.0f / (float)NPTS;
    const float cx = sx * inv_n, cy = sy * inv_n, cz = sz * inv_n;

    // per-row squared norm + centroid distance
    for (int r = lane; r < 64; r += 32) {
        float x = P[r * 4 + 0], y = P[r * 4 + 1], z = P[r * 4 + 2];
        SQ[r] = x * x + y * y + z * z;
        float dx = x - cx, dy = y - cy, dz = z - cz;
        FD[2 * r] = sqrtf(dx * dx + dy * dy + dz * dz);
    }
    __syncthreads();

    const int  lm = lane & 15;
    const bool hi = lane >= 16;
    const int  rowbase = hi ? 8 : 0;

    // A/B fragment for tile t per the 16x4 f32 WMMA layout:
    // lanes 0-15 hold (K0,K1)=(x,y); lanes 16-31 hold (K2,K3)=(z,0)
    v2f frag[4];
    float sqcol[4];
    #pragma unroll
    for (int t = 0; t < 4; ++t) {
        int r = t * 16 + lm;
        float f0 = hi ? P[r * 4 + 2] : P[r * 4 + 0];
        float f1 = hi ? 0.0f         : P[r * 4 + 1];
        frag[t] = (v2f){f0, f1};
        sqcol[t] = SQ[r];
    }

    #pragma unroll
    for (int ti = 0; ti < 4; ++ti) {
        v8f acc[4];
        #pragma unroll
        for (int tj = 0; tj < 4; ++tj) {
#if USE_WMMA
            v8f z = {};
            acc[tj] = __builtin_amdgcn_wmma_f32_16x16x4_f32(
                false, frag[ti], false, frag[tj], (short)0, z, false, false);
#else
            #pragma unroll
            for (int k = 0; k < 8; ++k) {
                int i = ti * 16 + k + rowbase;
                int j = tj * 16 + lm;
                acc[tj][k] = P[i*4+0]*P[j*4+0] + P[i*4+1]*P[j*4+1] + P[i*4+2]*P[j*4+2];
            }
#endif
        }
        // fused store:  G'[i][j] = SQ[j] - 2*ip   (conflict-free pattern)
        #pragma unroll
        for (int tj = 0; tj < 4; ++tj) {
            #pragma unroll
            for (int k = 0; k < 8; ++k) {
                int i = ti * 16 + k + rowbase;
                int j = tj * 16 + lm;
                G[i * GS + j] = fmaf(-2.0f, acc[tj][k], sqcol[tj]);
            }
        }
    }

    // poison diagonal and columns 49..51 (removes all masking from the scan)
    G[lane * GS + lane]               = POIS;
    G[(lane + 32) * GS + (lane + 32)] = POIS;
    #pragma unroll
    for (int t = 0; t < 6; ++t) {
        int idx = t * 32 + lane;          // 0..191
        int r = idx / 3;
        int c = 49 + (idx - r * 3);
        G[r * GS + c] = POIS;
    }
    __syncthreads();

    // each lane scans rows (lane) and (lane+32): sorted-triple insert, 5 ops/value
    const int rA = lane, rB = lane + 32;
    const float sqA = SQ[rA], sqB = SQ[rB];
    float a0 = BIGF, a1 = BIGF, a2 = BIGF;
    float e0 = BIGF, e1 = BIGF, e2 = BIGF;
    const v4f* GA = (const v4f*)(G + rA * GS);
    const v4f* GB = (const v4f*)(G + rB * GS);
    #pragma unroll
    for (int g4 = 0; g4 < 13; ++g4) {        // columns 0..51 (49..51 poisoned)
        v4f ga = GA[g4];
        v4f gb = GB[g4];
        #pragma unroll
        for (int c = 0; c < 4; ++c) {
            float xA = sqA + ga[c];
            float tA = fmaxf(a0, xA); a0 = fminf(a0, xA);
            float uA = fmaxf(a1, tA); a1 = fminf(a1, tA);
            a2 = fminf(a2, uA);
            float xB = sqB + gb[c];
            float tB = fmaxf(e0, xB); e0 = fminf(e0, xB);
            float uB = fmaxf(e1, tB); e1 = fminf(e1, tB);
            e2 = fminf(e2, uB);
        }
    }
    a0 = fmaxf(a0, 0.0f); a1 = fmaxf(a1, 0.0f); a2 = fmaxf(a2, 0.0f);
    e0 = fmaxf(e0, 0.0f); e1 = fmaxf(e1, 0.0f); e2 = fmaxf(e2, 0.0f);
    const float third = 1.0f / 3.0f;
    FD[2 * rA + 1] = (sqrtf(a0) + sqrtf(a1) + sqrtf(a2)) * third;
    if (rB < NPTS)
        FD[2 * rB + 1] = (sqrtf(e0) + sqrtf(e1) + sqrtf(e2)) * third;
    __syncthreads();

    // streamed rank-2 output; lane owns 4 contiguous channels
    const v4f uu = ((const v4f*)(uvc))[lane];
    const v4f vv = ((const v4f*)(uvc + EOUT))[lane];
    const v4f cc = ((const v4f*)(uvc + 2 * EOUT))[lane];
    float* ob = out + (size_t)b * NPTS * EOUT;
    #pragma unroll 7
    for (int r = 0; r < NPTS; ++r) {
        v2f fd = ((const v2f*)FD)[r];
        float cd = fd.x;
        float dn = fd.y;
        v4f o;
        o.x = fmaf(cd, uu.x, fmaf(dn, vv.x, cc.x));
        o.y = fmaf(cd, uu.y, fmaf(dn, vv.y, cc.y));
        o.z = fmaf(cd, uu.z, fmaf(dn, vv.z, cc.z));
        o.w = fmaf(cd, uu.w, fmaf(dn, vv.w, cc.w));
        __builtin_nontemporal_store(o, (v4f*)(ob + r * EOUT) + lane);
    }
}

// ---------------------------------------------------------------------------
extern "C" void kernel_launch(void* const* d_in, const int* in_sizes, int n_in,
                              void* d_out, int out_size, void* d_ws, size_t ws_size,
                              hipStream_t stream)
{
    const float* points = (const float*)d_in[0];
    const float* W_dist = (const float*)d_in[1];
    const float* b_dist = (const float*)d_in[2];
    const float* emb    = (const float*)d_in[3];
    const float* W_den  = (const float*)d_in[4];
    const float* b_den  = (const float*)d_in[5];
    const float* W_out  = (const float*)d_in[6];
    const float* b_out  = (const float*)d_in[7];
    float* out = (float*)d_out;
    float* uvc = (float*)d_ws;   // 3*128 floats

    const int B = in_sizes[0] / (NPTS * 3);

    uvc_kernel<<<1, EOUT, 0, stream>>>(W_dist, b_dist, emb, W_den, b_den,
                                       W_out, b_out, uvc);
    fused_kernel<<<B, 32, 0, stream>>>(points, uvc, out);
}